// GAT_84284438216995
// MI455X (gfx1250) — compile-verified
//
#include <hip/hip_runtime.h>
#include <hip/hip_bf16.h>
#include <math.h>

typedef __attribute__((ext_vector_type(2))) float v2f;
typedef __attribute__((ext_vector_type(8))) float v8f;

// ---------------------------------------------------------------------------
// fill (grid-stride)
// ---------------------------------------------------------------------------
__global__ void fill_kernel(float* __restrict__ p, float v, size_t n) {
  size_t i = (size_t)blockIdx.x * blockDim.x + threadIdx.x;
  size_t stride = (size_t)gridDim.x * blockDim.x;
  for (; i < n; i += stride) p[i] = v;
}

// ---------------------------------------------------------------------------
// GEMM: C[M,Nout] = A[M,K] @ B[K,Nout]   (row-major), f32 WMMA 16x16x4.
// Block = 256 threads = 8 waves; tile 32(M) x 64(N); waves laid out 2x4.
// Tiles staged into LDS with CDNA5 GLOBAL_LOAD_ASYNC_TO_LDS_B128 (ASYNCcnt),
// GVS addressing: SGPR base + per-lane 32-bit byte offset.
// ---------------------------------------------------------------------------
#define BM 32
#define BN 64
#define KT 32

__global__ __launch_bounds__(256) void gemm_wmma(
    const float* __restrict__ A, const float* __restrict__ B,
    float* __restrict__ C, int M, int K, int Nout) {
  // KT+4 pad keeps every 4-float group 16B-aligned for async b128 LDS writes
  // (stride 36 floats; column reads are at worst 2-way bank conflicted).
  __shared__ float As[BM][KT + 4];
  __shared__ float Bs[KT][BN];

  int tn_count = (Nout + BN - 1) / BN;
  int btm = blockIdx.x / tn_count;
  int btn = blockIdx.x % tn_count;
  int r0 = btm * BM;
  int n0 = btn * BN;

  int t = threadIdx.x;
  int lane = t & 31;
  int wid  = t >> 5;
  int wm = wid & 1;        // 0..1  (M direction, 16 rows each)
  int wn = wid >> 1;       // 0..3  (N direction, 16 cols each)

  // staging assignments (constant over k-chunks)
  int row_a = t >> 3;              // 0..31
  int cg_a  = (t & 7) << 2;        // 0,4,...,28
  int gr    = r0 + row_a;
  unsigned lds_a = (unsigned)(size_t)&As[row_a][cg_a];

  int row_b = t >> 4;              // 0..15
  int cg_b  = (t & 15) << 2;       // 0..60
  int gc    = n0 + cg_b;
  bool b_in = (gc + 3) < Nout;     // Nout % 4 == 0 -> group fully in or out
  unsigned lds_b0 = (unsigned)(size_t)&Bs[row_b][cg_b];
  unsigned lds_b1 = (unsigned)(size_t)&Bs[row_b + 16][cg_b];

  v8f acc = {};

  for (int k0 = 0; k0 < K; k0 += KT) {
    // ---- stage A tile 32x32: one async b128 per thread ----
    if (gr < M) {
      unsigned voff = ((unsigned)gr * (unsigned)K + (unsigned)(k0 + cg_a)) * 4u;
      asm volatile("global_load_async_to_lds_b128 %0, %1, %2"
                   :: "v"(lds_a), "v"(voff), "s"(A) : "memory");
    } else {
      As[row_a][cg_a + 0] = 0.f; As[row_a][cg_a + 1] = 0.f;
      As[row_a][cg_a + 2] = 0.f; As[row_a][cg_a + 3] = 0.f;
    }
    // ---- stage B tile 32x64: two async b128 per thread ----
    if (b_in) {
      unsigned voff0 = ((unsigned)(k0 + row_b)      * (unsigned)Nout + (unsigned)gc) * 4u;
      unsigned voff1 = ((unsigned)(k0 + row_b + 16) * (unsigned)Nout + (unsigned)gc) * 4u;
      asm volatile("global_load_async_to_lds_b128 %0, %1, %2"
                   :: "v"(lds_b0), "v"(voff0), "s"(B) : "memory");
      asm volatile("global_load_async_to_lds_b128 %0, %1, %2"
                   :: "v"(lds_b1), "v"(voff1), "s"(B) : "memory");
    } else {
      for (int j = 0; j < 4; ++j) {
        Bs[row_b][cg_b + j]      = 0.f;
        Bs[row_b + 16][cg_b + j] = 0.f;
      }
    }
    // async LDS writes must land before DS reads; __syncthreads only waits DScnt
    asm volatile("s_wait_asynccnt 0x0" ::: "memory");
    __syncthreads();

    // ISA 7.12.2: A 16x4 f32: lane m=l&15, lanes>=16 carry K+2..K+3.
    //            B 4x16 f32: lane n=l&15, VGPR v holds row K = v + 2*(l>>4).
    int mloc = wm * 16 + (lane & 15);
    int nloc = wn * 16 + (lane & 15);
    int kb   = (lane >> 4) << 1;

    #pragma unroll
    for (int kk = 0; kk < KT; kk += 4) {
      v2f af, bf;
      af.x = As[mloc][kk + kb];
      af.y = As[mloc][kk + kb + 1];
      bf.x = Bs[kk + kb][nloc];
      bf.y = Bs[kk + kb + 1][nloc];
      acc = __builtin_amdgcn_wmma_f32_16x16x4_f32(
          false, af, false, bf, (short)0, acc, false, false);
    }
    __syncthreads();
  }

  // C/D layout: lane l, VGPR v -> row = v + 8*(l>>4), col = l&15
  int col = n0 + wn * 16 + (lane & 15);
  if (col < Nout) {
    int rbase = r0 + wm * 16 + ((lane >> 4) << 3);
    for (int v = 0; v < 8; ++v) {
      int row = rbase + v;
      if (row < M) C[(size_t)row * Nout + col] = acc[v];
    }
  }
}

// ---------------------------------------------------------------------------
// attention logits, 8 heads x 32 dims: wave per node
// ---------------------------------------------------------------------------
__global__ __launch_bounds__(256) void att_logits8(
    const float* __restrict__ h, const float* __restrict__ a_s,
    const float* __restrict__ a_d, float* __restrict__ alS,
    float* __restrict__ alD, int N) {
  int n = blockIdx.x * (blockDim.x >> 5) + (threadIdx.x >> 5);
  if (n >= N) return;
  int lane = threadIdx.x & 31;
  const float* hp = h + (size_t)n * 256 + lane * 8;
  float s = 0.f, d = 0.f;
  #pragma unroll
  for (int j = 0; j < 8; ++j) {
    float hv = hp[j];
    s += hv * a_s[lane * 8 + j];   // head=(l>>2), dim=(l&3)*8+j  ->  l*8+j
    d += hv * a_d[lane * 8 + j];
  }
  s += __shfl_xor(s, 1); s += __shfl_xor(s, 2);
  d += __shfl_xor(d, 1); d += __shfl_xor(d, 2);
  if ((lane & 3) == 0) {
    int head = lane >> 2;
    alS[(size_t)n * 8 + head] = s;
    alD[(size_t)n * 8 + head] = d;
  }
}

// attention logits, 1 head x D dims: thread per node
__global__ void att_logits1(const float* __restrict__ h,
                            const float* __restrict__ a_s,
                            const float* __restrict__ a_d,
                            float* __restrict__ alS, float* __restrict__ alD,
                            int N, int D) {
  int n = blockIdx.x * blockDim.x + threadIdx.x;
  if (n >= N) return;
  float s = 0.f, d = 0.f;
  for (int j = 0; j < D; ++j) {
    float hv = h[(size_t)n * D + j];
    s += hv * a_s[j];
    d += hv * a_d[j];
  }
  alS[n] = s;
  alD[n] = d;
}

// ---------------------------------------------------------------------------
// float atomic max via signed-max / unsigned-min bit trick (init = -inf)
// ---------------------------------------------------------------------------
__device__ inline void atomicMaxF(float* addr, float val) {
  if (val >= 0.f) atomicMax((int*)addr, __float_as_int(val));
  else            atomicMin((unsigned int*)addr, __float_as_uint(val));
}

// pass A: per-(edge,head) leaky-relu logit, atomic max over destination
template <int H>
__global__ void edge_max(const int* __restrict__ src, const int* __restrict__ dst,
                         int E, int N,
                         const float* __restrict__ alS,
                         const float* __restrict__ alD,
                         float* __restrict__ maxb) {
  int idx = blockIdx.x * blockDim.x + threadIdx.x;
  int ET = E + N;
  if (idx >= ET * H) return;
  int e = idx / H, hd = idx % H;   // H is power of 2 -> shifts
  int s, d;
  if (e < E) { s = src[e]; d = dst[e]; } else { s = e - E; d = s; }
  float x = alS[(size_t)s * H + hd] + alD[(size_t)d * H + hd];
  x = x > 0.f ? x : 0.2f * x;
  atomicMaxF(&maxb[(size_t)d * H + hd], x);
}

// pass B (H=8, D=32): wave per edge, coalesced 1KB row gather + atomic scatter
__global__ __launch_bounds__(256) void edge_accum256(
    const int* __restrict__ src, const int* __restrict__ dst, int E, int N,
    const float* __restrict__ h, const float* __restrict__ alS,
    const float* __restrict__ alD, const float* __restrict__ maxb,
    float* __restrict__ sumb, float* __restrict__ accum) {
  int e = blockIdx.x * (blockDim.x >> 5) + (threadIdx.x >> 5);
  int ET = E + N;
  if (e >= ET) return;
  int lane = threadIdx.x & 31;
  int s, d;
  if (e < E) { s = src[e]; d = dst[e]; } else { s = e - E; d = s; }
  int head = lane >> 2;
  float lg = alS[(size_t)s * 8 + head] + alD[(size_t)d * 8 + head];
  lg = lg > 0.f ? lg : 0.2f * lg;
  float w = expf(lg - maxb[(size_t)d * 8 + head]);
  if ((lane & 3) == 0) atomicAdd(&sumb[(size_t)d * 8 + head], w);
  const float* hp = h + (size_t)s * 256 + lane * 8;
  float* ap = accum + (size_t)d * 256 + lane * 8;
  float4 h0 = *(const float4*)hp;
  float4 h1 = *(const float4*)(hp + 4);
  atomicAdd(ap + 0, w * h0.x); atomicAdd(ap + 1, w * h0.y);
  atomicAdd(ap + 2, w * h0.z); atomicAdd(ap + 3, w * h0.w);
  atomicAdd(ap + 4, w * h1.x); atomicAdd(ap + 5, w * h1.y);
  atomicAdd(ap + 6, w * h1.z); atomicAdd(ap + 7, w * h1.w);
}

// pass B (H=1, D=40): wave per edge, lanes 0..19 carry 2 dims each
__global__ __launch_bounds__(256) void edge_accum40(
    const int* __restrict__ src, const int* __restrict__ dst, int E, int N,
    const float* __restrict__ h, const float* __restrict__ alS,
    const float* __restrict__ alD, const float* __restrict__ maxb,
    float* __restrict__ sumb, float* __restrict__ accum) {
  int e = blockIdx.x * (blockDim.x >> 5) + (threadIdx.x >> 5);
  int ET = E + N;
  if (e >= ET) return;
  int lane = threadIdx.x & 31;
  int s, d;
  if (e < E) { s = src[e]; d = dst[e]; } else { s = e - E; d = s; }
  float lg = alS[s] + alD[d];
  lg = lg > 0.f ? lg : 0.2f * lg;
  float w = expf(lg - maxb[d]);
  if (lane == 0) atomicAdd(&sumb[d], w);
  if (lane < 20) {
    float2 hv = *(const float2*)(h + (size_t)s * 40 + lane * 2);
    atomicAdd(accum + (size_t)d * 40 + lane * 2 + 0, w * hv.x);
    atomicAdd(accum + (size_t)d * 40 + lane * 2 + 1, w * hv.y);
  }
}

// finalize layers 1/2: divide by softmax sum, add bias, ELU
__global__ void finalize_elu(const float* __restrict__ accum,
                             const float* __restrict__ sumb,
                             const float* __restrict__ bias,
                             float* __restrict__ out, int N) {
  int idx = blockIdx.x * blockDim.x + threadIdx.x;
  if (idx >= N * 256) return;
  int n = idx >> 8, i = idx & 255;
  float v = accum[idx] / (sumb[(size_t)n * 8 + (i >> 5)] + 1e-16f) + bias[i];
  out[idx] = v > 0.f ? v : expm1f(v);
}

// finalize layer 3: divide, add bias (no ELU), D=40
__global__ void finalize3(const float* __restrict__ accum,
                          const float* __restrict__ sumb,
                          const float* __restrict__ bias,
                          float* __restrict__ out, int N) {
  int idx = blockIdx.x * blockDim.x + threadIdx.x;
  if (idx >= N * 40) return;
  int n = idx / 40, i = idx % 40;
  out[idx] = accum[idx] / (sumb[n] + 1e-16f) + bias[i];
}

// ---------------------------------------------------------------------------
extern "C" void kernel_launch(void* const* d_in, const int* in_sizes, int n_in,
                              void* d_out, int out_size, void* d_ws, size_t ws_size,
                              hipStream_t stream) {
  const float* x   = (const float*)d_in[0];
  const int*   ei  = (const int*)d_in[1];
  const float* W1  = (const float*)d_in[2];
  const float* as1 = (const float*)d_in[3];
  const float* ad1 = (const float*)d_in[4];
  const float* b1  = (const float*)d_in[5];
  const float* W2  = (const float*)d_in[6];
  const float* as2 = (const float*)d_in[7];
  const float* ad2 = (const float*)d_in[8];
  const float* b2  = (const float*)d_in[9];
  const float* W3  = (const float*)d_in[10];
  const float* as3 = (const float*)d_in[11];
  const float* ad3 = (const float*)d_in[12];
  const float* b3  = (const float*)d_in[13];

  const int N  = in_sizes[0] / 128;
  const int E  = in_sizes[1] / 2;
  const int ET = E + N;
  const int* src = ei;
  const int* dst = ei + E;

  float* ws    = (float*)d_ws;
  float* hW    = ws;                          // [N,256] GEMM output
  float* accum = hW    + (size_t)N * 256;     // [N,256] scatter accumulator
  float* hbuf  = accum + (size_t)N * 256;     // [N,256] layer activations
  float* alS   = hbuf  + (size_t)N * 256;     // [N,8]
  float* alD   = alS   + (size_t)N * 8;       // [N,8]
  float* maxb  = alD   + (size_t)N * 8;       // [N,8]
  float* sumb  = maxb  + (size_t)N * 8;       // [N,8]

  dim3 blk(256);
  const int tm = (N + BM - 1) / BM;

  // ---------------- layer 1: 128 -> 8x32 ----------------
  gemm_wmma<<<dim3(tm * 4), blk, 0, stream>>>(x, W1, hW, N, 128, 256);
  att_logits8<<<dim3((N + 7) / 8), blk, 0, stream>>>(hW, as1, ad1, alS, alD, N);
  fill_kernel<<<dim3(1024), blk, 0, stream>>>(maxb, -INFINITY, (size_t)N * 8);
  fill_kernel<<<dim3(1024), blk, 0, stream>>>(sumb, 0.f, (size_t)N * 8);
  fill_kernel<<<dim3(8192), blk, 0, stream>>>(accum, 0.f, (size_t)N * 256);
  edge_max<8><<<dim3((ET * 8 + 255) / 256), blk, 0, stream>>>(src, dst, E, N, alS, alD, maxb);
  edge_accum256<<<dim3((ET + 7) / 8), blk, 0, stream>>>(src, dst, E, N, hW, alS, alD, maxb, sumb, accum);
  finalize_elu<<<dim3((N * 256 + 255) / 256), blk, 0, stream>>>(accum, sumb, b1, hbuf, N);

  // ---------------- layer 2: 256 -> 8x32 ----------------
  gemm_wmma<<<dim3(tm * 4), blk, 0, stream>>>(hbuf, W2, hW, N, 256, 256);
  att_logits8<<<dim3((N + 7) / 8), blk, 0, stream>>>(hW, as2, ad2, alS, alD, N);
  fill_kernel<<<dim3(1024), blk, 0, stream>>>(maxb, -INFINITY, (size_t)N * 8);
  fill_kernel<<<dim3(1024), blk, 0, stream>>>(sumb, 0.f, (size_t)N * 8);
  fill_kernel<<<dim3(8192), blk, 0, stream>>>(accum, 0.f, (size_t)N * 256);
  edge_max<8><<<dim3((ET * 8 + 255) / 256), blk, 0, stream>>>(src, dst, E, N, alS, alD, maxb);
  edge_accum256<<<dim3((ET + 7) / 8), blk, 0, stream>>>(src, dst, E, N, hW, alS, alD, maxb, sumb, accum);
  finalize_elu<<<dim3((N * 256 + 255) / 256), blk, 0, stream>>>(accum, sumb, b2, hbuf, N);

  // ---------------- layer 3: 256 -> 40, 1 head ----------------
  gemm_wmma<<<dim3(tm * 1), blk, 0, stream>>>(hbuf, W3, hW, N, 256, 40);
  att_logits1<<<dim3((N + 255) / 256), blk, 0, stream>>>(hW, as3, ad3, alS, alD, N, 40);
  fill_kernel<<<dim3(256), blk, 0, stream>>>(maxb, -INFINITY, (size_t)N);
  fill_kernel<<<dim3(256), blk, 0, stream>>>(sumb, 0.f, (size_t)N);
  fill_kernel<<<dim3(2048), blk, 0, stream>>>(accum, 0.f, (size_t)N * 40);
  edge_max<1><<<dim3((ET + 255) / 256), blk, 0, stream>>>(src, dst, E, N, alS, alD, maxb);
  edge_accum40<<<dim3((ET + 7) / 8), blk, 0, stream>>>(src, dst, E, N, hW, alS, alD, maxb, sumb, accum);
  finalize3<<<dim3((N * 40 + 255) / 256), blk, 0, stream>>>(accum, sumb, b3, (float*)d_out, N);
}